// SoftDTWTriton_26448408609461
// MI455X (gfx1250) — compile-verified
//
#include <hip/hip_runtime.h>

typedef __attribute__((ext_vector_type(2))) float v2f;
typedef __attribute__((ext_vector_type(8))) float v8f;

#define BATCH 32
#define NN 512
#define MM 512
#define KD 64
#define NDIAG (NN + MM - 1)

// ---------------------------------------------------------------------------
// Phase 1: D[b, p, i] = ||X[b,i]||^2 + ||Y[b,j]||^2 - 2 * X[b,i]·Y[b,j]
// stored anti-diagonal-major (p = i + j) so the wavefront kernels coalesce.
// One wave computes one 16x16 tile via f32 WMMA (16 steps of K=4).
// Row norms are recovered from the already-loaded fragments via shuffles.
// ---------------------------------------------------------------------------
__global__ void __launch_bounds__(256)
sqdist_wmma_kernel(const float* __restrict__ X, const float* __restrict__ Y,
                   float* __restrict__ Dd) {
  const int b    = blockIdx.x;
  const int wave = threadIdx.x >> 5;
  const int lane = threadIdx.x & 31;
  const int l    = lane & 15;   // column-within-tile / A-row lane
  const int h    = lane >> 4;   // half select (K pair for A/B frags)
  const int tileM = blockIdx.y * 16;
  const int tileN = (blockIdx.z * 8 + wave) * 16;

  // A frag lane (h,l): X[tileM+l][k + 2h .. +1]   (16x4 f32 A layout)
  // B frag lane (h,l): Y[tileN+l][k + 2h .. +1]   (4x16 f32 B layout, B[k][n]=Y[n][k])
  const float* Xr = X + ((size_t)b * NN + tileM + l) * KD + 2 * h;
  const float* Yr = Y + ((size_t)b * MM + tileN + l) * KD + 2 * h;

  v8f c = {};
  float xpart = 0.f, ypart = 0.f;
#pragma unroll
  for (int k = 0; k < KD; k += 4) {
    v2f a  = *(const v2f*)(Xr + k);
    v2f bb = *(const v2f*)(Yr + k);
    xpart += a.x * a.x + a.y * a.y;
    ypart += bb.x * bb.x + bb.y * bb.y;
    // D = A*B + C, f32, 16x16x4
    c = __builtin_amdgcn_wmma_f32_16x16x4_f32(false, a, false, bb, (short)0, c,
                                              false, false);
  }
  // each lane holds half of the squared-norm of its row; fold halves
  float xx = xpart + __shfl_xor(xpart, 16);  // ||X[tileM+l]||^2 (lanes l, l+16)
  float yy = ypart + __shfl_xor(ypart, 16);  // ||Y[tileN+l]||^2

  float* Db = Dd + (size_t)b * NDIAG * NN;
#pragma unroll
  for (int r = 0; r < 8; ++r) {
    // C/D layout: VGPR r, lane (h,l) -> element [M = r + 8h][N = l]
    int m = tileM + r + 8 * h;
    int n = tileN + l;
    float xv = __shfl(xx, r + 8 * h);  // norm of row m
    Db[(size_t)(m + n) * NN + m] = xv + yy - 2.0f * c[r];
  }
}

__device__ __forceinline__ void band_limits(int p, int bandwidth, int& i_start,
                                            int& i_end) {
  i_start = max(0, p - (MM - 1));
  i_end   = min(p, NN - 1);
  if (bandwidth > 0) {
    i_start = max(i_start, (p - bandwidth + 1) >> 1);  // arithmetic shift == floor-div
    i_end   = min(i_end, (p + bandwidth) >> 1);
  }
}

// ---------------------------------------------------------------------------
// Phase 2: forward softmin wavefront. One block per batch, 512 threads.
// Rolling 3-diagonal LDS buffers in padded coords: buf[q%3][s] = Rpad[s][q-s].
// Thread t owns slot t+1 of the new diagonal (exactly one writer per slot ->
// a single barrier per step). Computed cells stream to global R (diag-major).
// gamma = 1.0 (reference GAMMA).
// ---------------------------------------------------------------------------
__global__ void __launch_bounds__(NN)
softdtw_fwd_kernel(const float* __restrict__ Dd, float* __restrict__ Rd,
                   const int* __restrict__ bw_ptr) {
  const int b = blockIdx.x;
  const int t = threadIdx.x;
  const int bandwidth = *bw_ptr;
  const float INF = __builtin_inff();

  __shared__ float buf[3][NN + 2];
  for (int s = t; s < NN + 2; s += NN) {
    buf[0][s] = (s == 0) ? 0.0f : INF;  // diag 0: Rpad[0][0] = 0
    buf[1][s] = INF;                    // diag 1: all INF
  }
  __syncthreads();

  const float* Db = Dd + (size_t)b * NDIAG * NN;
  float* Rb = Rd + (size_t)b * NDIAG * NN;

  for (int p = 0; p < NDIAG; ++p) {
    float* b0 = buf[p % 3];        // padded diag p   (reads Rpad[i][j])
    float* b1 = buf[(p + 1) % 3];  // padded diag p+1 (reads Rpad[i][j+1], Rpad[i+1][j])
    float* b2 = buf[(p + 2) % 3];  // padded diag p+2 (writes Rpad[i+1][j+1])

    int i_start, i_end;
    band_limits(p, bandwidth, i_start, i_end);

    float out = INF;
    if (t >= i_start && t <= i_end) {
      float r00 = b0[t];      // Rpad[i][j]
      float r01 = b1[t];      // Rpad[i][j+1]
      float r10 = b1[t + 1];  // Rpad[i+1][j]
      float z0 = -r00, z1 = -r01, z2 = -r10;  // * inv_gamma (=1)
      float zmax = fmaxf(z0, fmaxf(z1, z2));
      float ssum = expf(z0 - zmax) + expf(z1 - zmax) + expf(z2 - zmax);
      float softmin = -(logf(ssum) + zmax);   // * gamma (=1)
      out = Db[(size_t)p * NN + t] + softmin; // coalesced diag-major load
      Rb[(size_t)p * NN + t] = out;           // coalesced diag-major store
    }
    b2[t + 1] = out;          // slot t+1 <-> interior row t of diag p+2
    if (t == 0) b2[0] = INF;  // padded boundary Rpad[0][p+2]
    __syncthreads();
  }
}

// ---------------------------------------------------------------------------
// Phase 3: backward E wavefront (reverse diagonals). E diagonals roll in LDS;
// R and D come from global (L2-resident streams, coalesced). Border reads of
// the padded arrays (-INF rows/cols, corner = Rpad[N][M]) are handled by
// guards instead of materializing padding. E interior goes straight to d_out.
// ---------------------------------------------------------------------------
__global__ void __launch_bounds__(NN)
softdtw_bwd_kernel(const float* __restrict__ Dd, const float* __restrict__ Rd,
                   float* __restrict__ E, const int* __restrict__ bw_ptr) {
  const int b = blockIdx.x;
  const int t = threadIdx.x;
  const int bandwidth = *bw_ptr;

  __shared__ float ebuf[3][NN];  // ebuf[q%3][i] = E_interior[i][q-i]

  const float* Db = Dd + (size_t)b * NDIAG * NN;
  const float* Rb = Rd + (size_t)b * NDIAG * NN;
  float* Eb = E + (size_t)b * NN * MM;
  const float corner = Rb[(size_t)(NDIAG - 1) * NN + (NN - 1)];  // Rpad[N][M]

  for (int p = NDIAG - 1; p >= 0; --p) {
    int i_start, i_end;
    band_limits(p, bandwidth, i_start, i_end);

    int s1, e1, s2, e2;  // band membership of target diagonals
    band_limits(p + 1, bandwidth, s1, e1);
    band_limits(p + 2, bandwidth, s2, e2);

    if (t >= i_start && t <= i_end) {
      const int i = t;
      const int j = p - i;
      float r00 = Rb[(size_t)p * NN + i];
      float acc = 0.0f;
      // (di,dj) = (0,1): target (i, j+1) on diag p+1
      if (j + 1 < MM && i >= s1 && i <= e1) {
        float rk = Rb[(size_t)(p + 1) * NN + i];
        float dk = Db[(size_t)(p + 1) * NN + i];
        acc += expf(rk - r00 - dk) * ebuf[(p + 1) % 3][i];
      }
      // (1,0): target (i+1, j) on diag p+1
      if (i + 1 < NN && i + 1 >= s1 && i + 1 <= e1) {
        float rk = Rb[(size_t)(p + 1) * NN + i + 1];
        float dk = Db[(size_t)(p + 1) * NN + i + 1];
        acc += expf(rk - r00 - dk) * ebuf[(p + 1) % 3][i + 1];
      }
      // (1,1): target (i+1, j+1) on diag p+2 (or the padded corner)
      if (i + 1 < NN && j + 1 < MM) {
        if (i + 1 >= s2 && i + 1 <= e2) {
          float rk = Rb[(size_t)(p + 2) * NN + i + 1];
          float dk = Db[(size_t)(p + 2) * NN + i + 1];
          acc += expf(rk - r00 - dk) * ebuf[(p + 2) % 3][i + 1];
        }
      } else if (i + 1 == NN && j + 1 == MM) {
        acc += expf(corner - r00);  // Dp=0, Epad[N+1][M+1]=1
      }
      ebuf[p % 3][i] = acc;
      Eb[(size_t)i * MM + j] = acc;  // row-major output
    } else if (t < NN) {
      int j = p - t;
      if (j >= 0 && j < MM) Eb[(size_t)t * MM + j] = 0.0f;  // out-of-band E = 0
    }
    __syncthreads();
  }
}

// ---------------------------------------------------------------------------
extern "C" void kernel_launch(void* const* d_in, const int* in_sizes, int n_in,
                              void* d_out, int out_size, void* d_ws,
                              size_t ws_size, hipStream_t stream) {
  (void)in_sizes; (void)n_in; (void)out_size; (void)ws_size;
  const float* X = (const float*)d_in[0];
  const float* Y = (const float*)d_in[1];
  const int* bw = (const int*)d_in[2];

  float* Dd = (float*)d_ws;                       // 32 * 1023 * 512 floats
  float* Rd = Dd + (size_t)BATCH * NDIAG * NN;    // 32 * 1023 * 512 floats

  dim3 g1(BATCH, NN / 16, MM / 128);
  sqdist_wmma_kernel<<<g1, 256, 0, stream>>>(X, Y, Dd);
  softdtw_fwd_kernel<<<BATCH, NN, 0, stream>>>(Dd, Rd, bw);
  softdtw_bwd_kernel<<<BATCH, NN, 0, stream>>>(Dd, Rd, (float*)d_out, bw);
}